// SoftTripletLoss_8641474200296
// MI455X (gfx1250) — compile-verified
//
#include <hip/hip_runtime.h>
#include <hip/hip_bf16.h>
#include <math.h>
#include <stdint.h>

typedef float v2f __attribute__((ext_vector_type(2)));
typedef float v8f __attribute__((ext_vector_type(8)));

#define MARGIN   0.3f
#define NEG_FILL -1e9f

// Fixed problem geometry (reference: N=8192, D=128)
constexpr int D      = 128;
constexpr int KS     = 32;            // 32 WMMA k-steps of K=4
constexpr int CB     = 32;            // columns staged per LDS chunk
constexpr int LSTR   = 132;           // padded LDS row stride (floats): 528B = 33*16B,
                                      // bank-conflict-free for ds_load_b64 fragment reads
constexpr int CHUNKF = CB * LSTR;     // 4224 floats per buffer (16,896 B); x2 buffers

// ---------------------------------------------------------------------------
// Kernel 1: L2-normalize rows (D = 128: one wave per row, float4 per lane)
// ---------------------------------------------------------------------------
__global__ __launch_bounds__(256)
void norm_rows_kernel(const float* __restrict__ feat, float* __restrict__ fout, int N) {
    const int lane = threadIdx.x & 31;
    const int row  = (int)((blockIdx.x * blockDim.x + threadIdx.x) >> 5);
    if (row >= N) return;
    const float4 v = ((const float4*)(feat + (size_t)row * D))[lane];
    float ss = v.x * v.x + v.y * v.y + v.z * v.z + v.w * v.w;
#pragma unroll
    for (int off = 16; off >= 1; off >>= 1) ss += __shfl_xor(ss, off, 32);
    const float inv = 1.0f / fmaxf(sqrtf(ss), 1e-12f);
    float4 o;
    o.x = v.x * inv; o.y = v.y * inv; o.z = v.z * inv; o.w = v.w * inv;
    ((float4*)(fout + (size_t)row * D))[lane] = o;
}

// ---------------------------------------------------------------------------
// Kernel 2: fused sim GEMM (fp32 WMMA 16x16x4) + masked row reductions.
//  - Each wave owns 16 rows; A fragment (16 x 128) resident in 64 VGPRs.
//  - Workgroup stages 32-column chunks of B in LDS via async global->LDS
//    (global_load_async_to_lds_b128, ASYNCcnt), double-buffered.
//  - Each wave accumulates TWO independent 16x16 tiles per chunk (interleaved
//    wmma chains) for ILP, reading B fragments from LDS (conflict-free).
// ---------------------------------------------------------------------------
__global__ __launch_bounds__(256)
void triplet_rows_kernel(const float* __restrict__ f, const int* __restrict__ labels,
                         float* __restrict__ row_loss, int N) {
    __shared__ float smB[2 * CHUNKF];

    const int tid      = threadIdx.x;
    const int lane     = tid & 31;
    const int wv       = tid >> 5;
    const int row_base = ((int)blockIdx.x * 8 + wv) * 16;
    const int m        = lane & 15;   // row/col within tile for this lane
    const int hi       = lane >> 4;   // half-wave select
    const int koff     = hi * 2;      // K-pair owned by this half-wave

    // ---- A fragments: rows row_base..row_base+15, all K, in registers ----
    v2f a[KS];
    const float* arow = f + (size_t)(row_base + m) * D;
#pragma unroll
    for (int ks = 0; ks < KS; ++ks)
        a[ks] = *(const v2f*)(arow + ks * 4 + koff);

    int lab_row[8];
#pragma unroll
    for (int v = 0; v < 8; ++v) lab_row[v] = labels[row_base + hi * 8 + v];

    float pos_sum[8], pos_cnt[8], neg_max[8];
#pragma unroll
    for (int v = 0; v < 8; ++v) {
        pos_sum[v] = 0.0f; pos_cnt[v] = 0.0f; neg_max[v] = NEG_FILL;
    }

    // ---- async-copy geometry: 1 x 16B piece per thread per chunk ----------
    // chunk = CB rows x 128 floats (contiguous in global); piece p = tid:
    //   row r = tid>>3, 16B slot s = tid&7
    const int r = tid >> 3;
    const int s = tid & 7;
    const uint32_t lds_piece0 = (uint32_t)(uintptr_t)(&smB[0])
                              + (uint32_t)(r * (LSTR * 4) + s * 16);
    const uint32_t g_row_off  = (uint32_t)(r * D * 4 + s * 16);  // within chunk

    const int NC = N / CB;   // 256 chunks

    // prologue: issue chunk 0 into buffer 0
    {
        uint32_t goff = (uint32_t)(0 * CB) * (uint32_t)(D * 4) + g_row_off;
        asm volatile("global_load_async_to_lds_b128 %0, %1, %2"
                     :: "v"(lds_piece0), "v"(goff), "s"(f) : "memory");
    }

    for (int ci = 0; ci < NC; ++ci) {
        const int cur = ci & 1;
        if (ci + 1 < NC) {
            // issue next chunk into the other buffer, then wait for our own
            // piece of chunk ci (async loads complete in order)
            uint32_t goff = (uint32_t)((ci + 1) * CB) * (uint32_t)(D * 4) + g_row_off;
            uint32_t ldsa = lds_piece0 + (uint32_t)((cur ^ 1) * CHUNKF * 4);
            asm volatile("global_load_async_to_lds_b128 %0, %1, %2"
                         :: "v"(ldsa), "v"(goff), "s"(f) : "memory");
            asm volatile("s_wait_asynccnt 0x1" ::: "memory");
        } else {
            asm volatile("s_wait_asynccnt 0x0" ::: "memory");
        }
        __syncthreads();   // chunk ci visible to all waves

        const float* bufp  = &smB[cur * CHUNKF];
        const int    cbase = ci * CB;
        const int    lab_c0 = labels[cbase + m];
        const int    lab_c1 = labels[cbase + 16 + m];

        // two independent accumulation chains (tiles [cbase, cbase+16) and [cbase+16, cbase+32))
        const float* b0p = bufp + (size_t)(0 * 16 + m) * LSTR + koff;
        const float* b1p = bufp + (size_t)(1 * 16 + m) * LSTR + koff;
        v8f c0 = {}, c1 = {};
#pragma unroll
        for (int ks = 0; ks < KS; ++ks) {
            v2f b0 = *(const v2f*)(b0p + ks * 4);
            v2f b1 = *(const v2f*)(b1p + ks * 4);
            c0 = __builtin_amdgcn_wmma_f32_16x16x4_f32(
                     false, a[ks], false, b0, (short)0, c0, false, false);
            c1 = __builtin_amdgcn_wmma_f32_16x16x4_f32(
                     false, a[ks], false, b1, (short)0, c1, false, false);
        }

#pragma unroll
        for (int v = 0; v < 8; ++v) {
            const int row = row_base + hi * 8 + v;
            {   // tile 0
                const int   col = cbase + m;
                const float sv  = c0[v];
                if (lab_row[v] == lab_c0) {
                    if (row != col) { pos_sum[v] += sv; pos_cnt[v] += 1.0f; }
                } else {
                    neg_max[v] = fmaxf(neg_max[v], sv);
                }
            }
            {   // tile 1
                const int   col = cbase + 16 + m;
                const float sv  = c1[v];
                if (lab_row[v] == lab_c1) {
                    if (row != col) { pos_sum[v] += sv; pos_cnt[v] += 1.0f; }
                } else {
                    neg_max[v] = fmaxf(neg_max[v], sv);
                }
            }
        }
        __syncthreads();   // all waves done reading before buffer is re-filled
    }

    // ---- reduce across the 16 lanes of each half-wave ---------------------
#pragma unroll
    for (int v = 0; v < 8; ++v) {
#pragma unroll
        for (int off = 8; off >= 1; off >>= 1) {
            pos_sum[v] += __shfl_xor(pos_sum[v], off, 32);
            pos_cnt[v] += __shfl_xor(pos_cnt[v], off, 32);
            neg_max[v]  = fmaxf(neg_max[v], __shfl_xor(neg_max[v], off, 32));
        }
    }

    if (m == 0) {
#pragma unroll
        for (int v = 0; v < 8; ++v) {
            const int   row = row_base + hi * 8 + v;
            const float ps  = (pos_cnt[v] > 0.0f) ? (pos_sum[v] / pos_cnt[v]) : 0.0f;
            const float ns  = (neg_max[v] > -1e8f) ? neg_max[v] : 0.0f;
            row_loss[row]   = fmaxf(0.0f, MARGIN + ns - ps);
        }
    }
}

// ---------------------------------------------------------------------------
// Kernel 3: mean over per-row losses -> scalar
// ---------------------------------------------------------------------------
__global__ __launch_bounds__(256)
void reduce_mean_kernel(const float* __restrict__ row_loss, float* __restrict__ out, int N) {
    __shared__ float sm[256];
    float ssum = 0.0f;
    for (int i = threadIdx.x; i < N; i += blockDim.x) ssum += row_loss[i];
    sm[threadIdx.x] = ssum;
    __syncthreads();
    for (int off = 128; off > 0; off >>= 1) {
        if ((int)threadIdx.x < off) sm[threadIdx.x] += sm[threadIdx.x + off];
        __syncthreads();
    }
    if (threadIdx.x == 0) out[0] = sm[0] / (float)N;
}

// ---------------------------------------------------------------------------
extern "C" void kernel_launch(void* const* d_in, const int* in_sizes, int n_in,
                              void* d_out, int out_size, void* d_ws, size_t ws_size,
                              hipStream_t stream) {
    const float* feat   = (const float*)d_in[0];
    const int*   labels = (const int*)d_in[1];
    const int    N      = in_sizes[1];          // 8192

    float* fnorm    = (float*)d_ws;             // N*128 floats (4 MB)
    float* row_loss = fnorm + (size_t)N * 128;  // N floats
    float* out      = (float*)d_out;

    // 1) normalize: 8 rows per 256-thread block (one wave per row)
    norm_rows_kernel<<<(N + 7) / 8, 256, 0, stream>>>(feat, fnorm, N);

    // 2) fused WMMA sim + masked reductions:
    //    8 waves/block * 16 rows/wave = 128 rows per block
    triplet_rows_kernel<<<N / 128, 256, 0, stream>>>(fnorm, labels, row_loss, N);

    // 3) scalar mean
    reduce_mean_kernel<<<1, 256, 0, stream>>>(row_loss, out, N);
}